// region_proposal_nets_81484119539873
// MI455X (gfx1250) — compile-verified
//
#include <hip/hip_runtime.h>
#include <math.h>

typedef __attribute__((ext_vector_type(2))) float v2f;
typedef __attribute__((ext_vector_type(8))) float v8f;
typedef int i32x4 __attribute__((vector_size(16)));

#define NIMG  4
#define CIN   512
#define COUT  512
#define HH    38
#define WW    38
#define HW    (HH * WW)          /* 1444  */
#define MTOT  (NIMG * HW)        /* 5776  */
#define NANCH (HW * 9)           /* 12996 */
#define NPRE  3000
#define NPOST 300
#define NSORT 16384
#define NMS_T 0.7f
#define MINSZ 16.0f

/* d_out float offsets (tuple: locs, scores, rois, rois_idx, anchor) */
#define OFF_LOCS   0
#define OFF_SCORES (MTOT * 36)                   /* 207936 */
#define OFF_ROIS   (OFF_SCORES + MTOT * 18)      /* 311904 */
#define OFF_RIDX   (OFF_ROIS + NIMG * NPOST * 4) /* 316704 */
#define OFF_ANCH   (OFF_RIDX + NIMG * NPOST)     /* 317904 */

/* workspace byte offsets */
#define WS_WP     0
#define WS_FEAT   (WS_WP + 9 * CIN * COUT * 4)
#define WS_BOXES  (WS_FEAT + MTOT * COUT * 4)
#define WS_KEYS   (WS_BOXES + NIMG * NANCH * 16)
#define WS_SBOX   (WS_KEYS + NIMG * NSORT * 8)
#define WS_SVAL   (WS_SBOX + NIMG * NPRE * 16)

/* ---------------- CDNA5 async global->LDS copy (probed via __has_builtin) - */
#if defined(__has_builtin)
# if __has_builtin(__builtin_amdgcn_global_load_async_to_lds_b32) && \
     __has_builtin(__builtin_amdgcn_global_load_async_to_lds_b128)
#  define HAVE_ASYNC_LDS 1
# endif
#endif

#ifdef HAVE_ASYNC_LDS
# define CP32(g, l)  __builtin_amdgcn_global_load_async_to_lds_b32( \
      (__attribute__((address_space(1))) int*)(g), \
      (__attribute__((address_space(3))) int*)(l), 0, 0)
# define CP128(g, l) __builtin_amdgcn_global_load_async_to_lds_b128( \
      (__attribute__((address_space(1))) i32x4*)(g), \
      (__attribute__((address_space(3))) i32x4*)(l), 0, 0)
#else
# define CP32(g, l)  (*(float*)(l) = *(const float*)(g))
# define CP128(g, l) (*(float4*)(l) = *(const float4*)(g))
#endif

__device__ __forceinline__ void wait_async_lds() {
#ifdef HAVE_ASYNC_LDS
# if __has_builtin(__builtin_amdgcn_s_wait_asynccnt)
    __builtin_amdgcn_s_wait_asynccnt(0);
# else
    asm volatile("s_wait_asynccnt 0" ::: "memory");
# endif
#endif
}

/* ---------------- 1. repack conv weights OIHW -> [kk][ci][co] ------------- */
__global__ void repack_w(const float* __restrict__ w, float* __restrict__ wp) {
    int t = blockIdx.x * blockDim.x + threadIdx.x;
    if (t >= COUT * CIN * 9) return;
    int co = t / (CIN * 9);
    int r  = t - co * (CIN * 9);
    int ci = r / 9;
    int kk = r - ci * 9;
    wp[(size_t)kk * CIN * COUT + (size_t)ci * COUT + co] = w[t];
}

/* ---------------- 2. 3x3 conv: LDS double-buffered implicit GEMM ---------- */
/* block tile M=32 x N=128, 8 waves, each wave: two 16x16 C tiles.           */
#define ASTRIDE 68                      /* 32 rows x 68 (pad) floats         */
#define BSTRIDE 132                     /* 64 rows x 132 (pad) floats        */
#define ACHUNK  (32 * ASTRIDE)          /* 2176 floats                       */
#define BCHUNK  (64 * BSTRIDE)          /* 8448 floats                       */
#define CONV_SMEM ((2 * ACHUNK + 2 * BCHUNK) * 4)   /* 84992 bytes           */
#define NCHUNK  72                      /* 9 taps x (512/64) ci-chunks       */

__global__ __launch_bounds__(256)
void conv3x3_wmma(const float* __restrict__ x, const float* __restrict__ wp,
                  const float* __restrict__ bias, float* __restrict__ feat) {
    extern __shared__ float smem[];
    float* Ab = smem;                    /* [2][32][ASTRIDE] */
    float* Bb = smem + 2 * ACHUNK;       /* [2][64][BSTRIDE] */

    const int tid     = threadIdx.x;
    const int blkX    = blockIdx.x;
    const int colBase = blockIdx.y * 128;

    /* staging mapping (fixed per thread) */
    const int amrow = tid & 31;                 /* A row it fills            */
    const int aciL  = tid >> 5;                 /* A ci slot base (+e*8)     */
    const int am    = blkX * 32 + amrow;
    const int an    = am / HW;
    const int arem  = am - an * HW;
    const int ay    = arem / WW;
    const int ax    = arem - ay * WW;
    const int bc16  = tid & 31;                 /* B 16B column slot         */
    const int brow0 = tid >> 5;                 /* B row base (+e*8)         */

    /* compute mapping */
    const int wave = tid >> 5, lane = tid & 31;
    const int l16  = lane & 15, hi = lane >> 4, koff = hi * 2;
    const int msub = wave >> 2;                 /* 0..1 -> M sub-tile        */
    const int csub = wave & 3;                  /* 0..3 -> 32-col sub-tile   */

    v8f acc0 = {}, acc1 = {};

    auto stage = [&](int c, int buf) {
        const int kk = c >> 3, ci0 = (c & 7) << 6;
        const int ky = kk / 3, kx = kk - ky * 3;
        const int iy = ay + ky - 1, ix = ax + kx - 1;
        const bool valid = (am < MTOT) && ((unsigned)iy < (unsigned)HH)
                                       && ((unsigned)ix < (unsigned)WW);
        const float* ag = x + (size_t)an * CIN * HW + (size_t)iy * WW + ix;
        float* Adst = Ab + buf * ACHUNK + amrow * ASTRIDE;
#pragma unroll
        for (int e = 0; e < 8; ++e) {
            const int cl = aciL + e * 8;
            if (valid) { CP32(ag + (size_t)(ci0 + cl) * HW, Adst + cl); }
            else       { Adst[cl] = 0.0f; }
        }
        const float* bg = wp + (size_t)kk * CIN * COUT + colBase + bc16 * 4;
        float* Bdst = Bb + buf * BCHUNK + bc16 * 4;
#pragma unroll
        for (int e = 0; e < 8; ++e) {
            const int row = brow0 + e * 8;
            CP128(bg + (size_t)(ci0 + row) * COUT, Bdst + row * BSTRIDE);
        }
    };

    auto compute = [&](int buf) {
        const float* Ar = Ab + buf * ACHUNK + (msub * 16 + l16) * ASTRIDE;
        const float* Br = Bb + buf * BCHUNK + csub * 32 + l16;
#pragma unroll
        for (int k = 0; k < 64; k += 4) {
            v2f a, b0, b1;
            a.x  = Ar[k + koff];
            a.y  = Ar[k + koff + 1];
            b0.x = Br[(k + koff) * BSTRIDE];
            b0.y = Br[(k + koff + 1) * BSTRIDE];
            b1.x = Br[(k + koff) * BSTRIDE + 16];
            b1.y = Br[(k + koff + 1) * BSTRIDE + 16];
            acc0 = __builtin_amdgcn_wmma_f32_16x16x4_f32(
                false, a, false, b0, (short)0, acc0, false, false);
            acc1 = __builtin_amdgcn_wmma_f32_16x16x4_f32(
                false, a, false, b1, (short)0, acc1, false, false);
        }
    };

    stage(0, 0);
    wait_async_lds();
    __syncthreads();
    for (int c = 0; c < NCHUNK; ++c) {
        const int cur = c & 1;
        if (c + 1 < NCHUNK) stage(c + 1, cur ^ 1);
        compute(cur);
        wait_async_lds();
        __syncthreads();
    }

    /* epilogue: bias + ReLU -> feat NHWC */
    const int   m16 = blkX * 32 + msub * 16;
    const int   col = colBase + csub * 32 + l16;
    const float bi0 = bias[col], bi1 = bias[col + 16];
#pragma unroll
    for (int r = 0; r < 8; ++r) {
        const int m = m16 + hi * 8 + r;
        if (m < MTOT) {
            float v0 = acc0[r] + bi0; v0 = v0 > 0.0f ? v0 : 0.0f;
            float v1 = acc1[r] + bi1; v1 = v1 > 0.0f ? v1 : 0.0f;
            feat[(size_t)m * COUT + col]      = v0;
            feat[(size_t)m * COUT + col + 16] = v1;
        }
    }
}

/* ---------------- 3. loc/score heads as one 54(->64)-col WMMA GEMM -------- */
__global__ void head_wmma(const float* __restrict__ feat,
                          const float* __restrict__ loc_w,
                          const float* __restrict__ loc_b,
                          const float* __restrict__ score_w,
                          const float* __restrict__ score_b,
                          float* __restrict__ out) {
    const int tid  = threadIdx.x;
    const int wave = tid >> 5;
    const int lane = tid & 31;
    const int l16  = lane & 15;
    const int hi   = lane >> 4;
    const int m0   = blockIdx.x * 16;
    const int col  = wave * 16 + l16;      /* 0..63; valid < 54 */
    const int koff = hi * 2;

    const float* __restrict__ wrow = nullptr;
    float bc = 0.0f;
    if (col < 36)      { wrow = loc_w   + (size_t)col * CIN;        bc = loc_b[col]; }
    else if (col < 54) { wrow = score_w + (size_t)(col - 36) * CIN; bc = score_b[col - 36]; }

    const float* __restrict__ abase = feat + (size_t)(m0 + l16) * CIN;
    v8f acc = {};
#pragma unroll 4
    for (int ci = 0; ci < CIN; ci += 4) {
        v2f a, b;
        a.x = abase[ci + koff];
        a.y = abase[ci + koff + 1];
        b.x = wrow ? wrow[ci + koff]     : 0.0f;
        b.y = wrow ? wrow[ci + koff + 1] : 0.0f;
        acc = __builtin_amdgcn_wmma_f32_16x16x4_f32(
            false, a, false, b, (short)0, acc, false, false);
    }
    const int rbase = m0 + hi * 8;
#pragma unroll
    for (int r = 0; r < 8; ++r) {
        float v = acc[r] + bc;
        int mrow = rbase + r;
        if (col < 36)
            out[OFF_LOCS + (size_t)mrow * 36 + col] = v;
        else if (col < 54)
            out[OFF_SCORES + (size_t)mrow * 18 + (col - 36)] = v;
    }
}

/* ---------------- 4. anchors --------------------------------------------- */
__global__ void anchors_k(float* __restrict__ out) {
    int t = blockIdx.x * blockDim.x + threadIdx.x;
    if (t >= NANCH) return;
    int a  = t % 9;
    int yx = t / 9;
    int y = yx / WW, x = yx - y * WW;
    const float ratios[3] = {0.5f, 1.0f, 2.0f};
    const float scales[3] = {8.0f, 16.0f, 32.0f};
    float r = ratios[a / 3], s = scales[a % 3];
    float hs = 16.0f * s * sqrtf(r);
    float ws = 16.0f * s * sqrtf(1.0f / r);
    float cx = x * 16.0f + 8.0f, cy = y * 16.0f + 8.0f;
    out[OFF_ANCH + (size_t)t * 4 + 0] = cx - ws * 0.5f;
    out[OFF_ANCH + (size_t)t * 4 + 1] = cy - hs * 0.5f;
    out[OFF_ANCH + (size_t)t * 4 + 2] = cx + ws * 0.5f;
    out[OFF_ANCH + (size_t)t * 4 + 3] = cy + hs * 0.5f;
}

/* ---------------- 5. decode boxes + build sort keys ----------------------- */
__global__ void decode_k(const float* __restrict__ out,
                         float4* __restrict__ boxes,
                         unsigned long long* __restrict__ keys,
                         const int* __restrict__ img_h,
                         const int* __restrict__ img_w) {
    int t = blockIdx.x * blockDim.x + threadIdx.x;
    if (t >= NIMG * NSORT) return;
    int n = t >> 14;
    int j = t & (NSORT - 1);
    unsigned long long key;
    if (j < NANCH) {
        int yx = j / 9, a = j - yx * 9;
        int m  = n * HW + yx;
        float s0 = out[OFF_SCORES + (size_t)m * 18 + 2 * a];
        float s1 = out[OFF_SCORES + (size_t)m * 18 + 2 * a + 1];
        float fg = 1.0f / (1.0f + expf(s0 - s1));          /* softmax fg prob */
        const float* loc = out + OFF_LOCS + (size_t)m * 36 + a * 4;
        const float* anc = out + OFF_ANCH + (size_t)j * 4;
        float aw  = anc[2] - anc[0], ah = anc[3] - anc[1];
        float acx = anc[0] + 0.5f * aw, acy = anc[1] + 0.5f * ah;
        float cx  = loc[0] * aw + acx, cy = loc[1] * ah + acy;
        float bw  = aw * expf(loc[2]), bh = ah * expf(loc[3]);
        float IW = (float)img_w[0], IH = (float)img_h[0];
        float x1 = fminf(fmaxf(cx - 0.5f * bw, 0.0f), IW);
        float y1 = fminf(fmaxf(cy - 0.5f * bh, 0.0f), IH);
        float x2 = fminf(fmaxf(cx + 0.5f * bw, 0.0f), IW);
        float y2 = fminf(fmaxf(cy + 0.5f * bh, 0.0f), IH);
        bool valid = ((x2 - x1) >= MINSZ) && ((y2 - y1) >= MINSZ);
        boxes[(size_t)n * NANCH + j] = make_float4(x1, y1, x2, y2);
        float ms = valid ? fg : -INFINITY;
        unsigned u = __float_as_uint(ms);
        u = (u & 0x80000000u) ? ~u : (u | 0x80000000u);    /* order-preserving */
        key = ((unsigned long long)u << 32) | (unsigned)(~(unsigned)j);
    } else {
        key = ((unsigned long long)0x007FFFFFu << 32) | (unsigned)(~(unsigned)j);
    }
    keys[t] = key;
}

/* ---------------- 6. per-image bitonic sort (desc) + top-3000 gather ------ */
__global__ void sort_k(const unsigned long long* __restrict__ keys,
                       const float4* __restrict__ boxes,
                       float4* __restrict__ sbox,
                       int* __restrict__ sval) {
    extern __shared__ unsigned long long sk[];  /* NSORT entries = 128KB LDS */
    const int n   = blockIdx.x;
    const int tid = threadIdx.x;
    for (int i = tid; i < NSORT; i += blockDim.x) sk[i] = keys[(size_t)n * NSORT + i];
    __syncthreads();
    for (int k = 2; k <= NSORT; k <<= 1) {
        for (int j = k >> 1; j > 0; j >>= 1) {
            for (int i = tid; i < NSORT; i += blockDim.x) {
                int ixj = i ^ j;
                if (ixj > i) {
                    bool desc = ((i & k) == 0);
                    unsigned long long a = sk[i], b = sk[ixj];
                    if ((a < b) == desc) { sk[i] = b; sk[ixj] = a; }
                }
            }
            __syncthreads();
        }
    }
    for (int i = tid; i < NPRE; i += blockDim.x) {
        unsigned long long key = sk[i];
        unsigned j = ~((unsigned)key);           /* recover index */
        unsigned u = (unsigned)(key >> 32);
        bool val = (j < (unsigned)NANCH) && (u != 0x007FFFFFu);
        float4 bb = (j < (unsigned)NANCH) ? boxes[(size_t)n * NANCH + j]
                                          : make_float4(0.f, 0.f, 0.f, 0.f);
        sbox[(size_t)n * NPRE + i] = bb;
        sval[(size_t)n * NPRE + i] = val ? 1 : 0;
    }
}

/* ---------------- 7. greedy NMS + cyclic select --------------------------- */
__global__ __launch_bounds__(1024)
void nms_k(const float4* __restrict__ sbox, const int* __restrict__ sval,
           float* __restrict__ out) {
    __shared__ float4        sb[NPRE];     /* 48000 B */
    __shared__ unsigned char kp[NPRE];     /*  3000 B */
    __shared__ short         lst[NPRE];    /*  6000 B */
    __shared__ int           nk;
    const int n   = blockIdx.x;
    const int tid = threadIdx.x;
    for (int i = tid; i < NPRE; i += blockDim.x) {
        sb[i] = sbox[(size_t)n * NPRE + i];
        kp[i] = (unsigned char)sval[(size_t)n * NPRE + i];
    }
    for (int i = 0; i < NPRE - 1; ++i) {
        __syncthreads();
        if (!kp[i]) continue;                  /* uniform across block */
        float4 bi = sb[i];
        float  ai = (bi.z - bi.x) * (bi.w - bi.y);
        for (int j = i + 1 + tid; j < NPRE; j += blockDim.x) {
            if (!kp[j]) continue;
            float4 bj = sb[j];
            float ix1 = fmaxf(bi.x, bj.x), iy1 = fmaxf(bi.y, bj.y);
            float ix2 = fminf(bi.z, bj.z), iy2 = fminf(bi.w, bj.w);
            float inter = fmaxf(ix2 - ix1, 0.0f) * fmaxf(iy2 - iy1, 0.0f);
            float uni   = ai + (bj.z - bj.x) * (bj.w - bj.y) - inter;
            float iou   = (uni > 0.0f) ? inter / uni : 0.0f;
            if (iou > NMS_T) kp[j] = 0;
        }
    }
    __syncthreads();
    if (tid == 0) {                            /* compact kept indices */
        int c = 0;
        for (int i = 0; i < NPRE; ++i)
            if (kp[i]) lst[c++] = (short)i;
        if (c == 0) { lst[0] = 0; c = 1; }
        nk = c;
    }
    __syncthreads();
    for (int p = tid; p < NPOST; p += blockDim.x) {
        float4 bb = sb[lst[p % nk]];
        size_t o = OFF_ROIS + ((size_t)n * NPOST + p) * 4;
        out[o + 0] = bb.x; out[o + 1] = bb.y; out[o + 2] = bb.z; out[o + 3] = bb.w;
        out[OFF_RIDX + (size_t)n * NPOST + p] = (float)n;
    }
}

/* ---------------- launch -------------------------------------------------- */
extern "C" void kernel_launch(void* const* d_in, const int* in_sizes, int n_in,
                              void* d_out, int out_size, void* d_ws, size_t ws_size,
                              hipStream_t stream) {
    const float* x       = (const float*)d_in[0];
    const float* conv1_w = (const float*)d_in[1];
    const float* conv1_b = (const float*)d_in[2];
    const float* score_w = (const float*)d_in[3];
    const float* score_b = (const float*)d_in[4];
    const float* loc_w   = (const float*)d_in[5];
    const float* loc_b   = (const float*)d_in[6];
    const int*   img_h   = (const int*)d_in[7];
    const int*   img_w   = (const int*)d_in[8];
    float* out = (float*)d_out;
    char*  ws  = (char*)d_ws;

    float*              wp    = (float*)(ws + WS_WP);
    float*              feat  = (float*)(ws + WS_FEAT);
    float4*             boxes = (float4*)(ws + WS_BOXES);
    unsigned long long* keys  = (unsigned long long*)(ws + WS_KEYS);
    float4*             sbox  = (float4*)(ws + WS_SBOX);
    int*                sval  = (int*)(ws + WS_SVAL);

    repack_w<<<(COUT * CIN * 9 + 255) / 256, 256, 0, stream>>>(conv1_w, wp);
    conv3x3_wmma<<<dim3((MTOT + 31) / 32, 4), 256, CONV_SMEM, stream>>>(x, wp, conv1_b, feat);
    head_wmma<<<MTOT / 16, 128, 0, stream>>>(feat, loc_w, loc_b, score_w, score_b, out);
    anchors_k<<<(NANCH + 255) / 256, 256, 0, stream>>>(out);
    decode_k<<<(NIMG * NSORT + 255) / 256, 256, 0, stream>>>(out, boxes, keys, img_h, img_w);
    sort_k<<<NIMG, 1024, NSORT * sizeof(unsigned long long), stream>>>(keys, boxes, sbox, sval);
    nms_k<<<NIMG, 1024, 0, stream>>>(sbox, sval, out);
}